// HierarchicalLLLoss_25967372272102
// MI455X (gfx1250) — compile-verified
//
#include <hip/hip_runtime.h>

// ---------------------------------------------------------------------------
// HierarchicalLLLoss on a balanced binary tree (C=1024 leaves, E=10 levels).
//
// Roofline: only `inputs` (B*C*4 = 32 MB) must be streamed; the one-hot slabs
// are implicit tree structure. At 23.3 TB/s this is ~1.4 us. The sums-of-4
// (levels 0->2, 75% of all adds) run on the WMMA pipe via
// V_WMMA_F32_16X16X4_F32 with an all-ones B matrix (B layout irrelevant),
// exact f32 arithmetic, A layout per ISA 7.12.2 (32-bit A 16x4).
//
// Round 2: software-pipeline stage 1 — issue all 32 A-operand loads up front
// (32 outstanding b64 loads per wave, LOADcnt<=63) so the compiler emits
// incremental s_wait_loadcnt instead of a full wait before every WMMA.
// ---------------------------------------------------------------------------

typedef __attribute__((ext_vector_type(2))) float v2f;
typedef __attribute__((ext_vector_type(8))) float v8f;

#define NCLS  1024        // C
#define NEDGE 10          // E
#define BATCH 8192        // B
#define SPB   16          // samples per block
#define NBLK  (BATCH / SPB)   // 512 blocks

__global__ __launch_bounds__(256) void hll_main(const float* __restrict__ inputs,
                                                const int*   __restrict__ target,
                                                const float* __restrict__ weights,
                                                float*       __restrict__ partial)
{
    __shared__ float bufA[256 * SPB];   // level-2 block sums g[q][m], 16 KB
    __shared__ float bufB[128 * SPB];   // fold ping-pong, 8 KB
    __shared__ float red[SPB];

    const int tid  = threadIdx.x;
    const int wave = tid >> 5;
    const int lane = tid & 31;
    const int m    = lane & 15;        // sample row (A-matrix M index)
    const int half = lane >> 4;        // 0 -> K=0,1 ; 1 -> K=2,3
    const int blk  = blockIdx.x;

    const float* row = inputs + (size_t)(blk * SPB + m) * NCLS;

    v2f ones; ones[0] = 1.0f; ones[1] = 1.0f;   // B = all ones (layout agnostic)

    // ---- Stage 1: levels 0->2 (sums of 4) on the matrix pipe --------------
    // Wave w owns 4-col chunks q in [32w, 32w+32): a 512-byte span per row.
    // Phase A: launch all 32 b64 loads (kept outstanding simultaneously).
    const float* base = row + wave * 128 + half * 2;   // wave*32 chunks * 4
    v2f a[32];
#pragma unroll
    for (int it = 0; it < 32; ++it)
        a[it] = *(const v2f*)(base + it * 4);

    // Phase B: drain in order: WMMA each chunk, scatter D[:,0] from lanes
    // 0 (rows 0..7) and 16 (rows 8..15) into LDS.
#pragma unroll
    for (int it = 0; it < 32; ++it) {
        const int q = wave * 32 + it;                  // chunk 0..255
        v8f c = {};
        v8f d = __builtin_amdgcn_wmma_f32_16x16x4_f32(
                    false, a[it], false, ones, (short)0, c, false, false);
        if (m == 0) {
            float* p = &bufA[q * SPB + half * 8];
            p[0] = d[0]; p[1] = d[1]; p[2] = d[2]; p[3] = d[3];
            p[4] = d[4]; p[5] = d[5]; p[6] = d[6]; p[7] = d[7];
        }
    }
    __syncthreads();

    // ---- Stage 2: fold levels 2 -> 10, capturing path siblings ------------
    int tgt = 0;
    if (tid < SPB) tgt = target[blk * SPB + tid];
    float sib[NEDGE];

    float* src = bufA;
    float* dst = bufB;
    int nblocks = 256;                     // block count at level 2
#pragma unroll
    for (int j = 2; j <= 9; ++j) {
        if (tid < SPB)                     // sibling of t's level-j block
            sib[j] = src[(((tgt >> j) ^ 1)) * SPB + tid];
        const int pairs = nblocks >> 1;
        for (int e = tid; e < pairs * SPB; e += 256) {
            const int k = e >> 4, mm = e & 15;
            dst[k * SPB + mm] = src[(2 * k) * SPB + mm]
                              + src[(2 * k + 1) * SPB + mm];
        }
        __syncthreads();
        float* tsw = src; src = dst; dst = tsw;
        nblocks = pairs;
    }
    // The s-chain below reconstructs the root sum as s9 + sib9, matching the
    // reference's den[E-1] exactly.

    // ---- Stage 3: fine path (levels 0,1) + log-chain per sample -----------
    if (tid < SPB) {
        const int t = tgt;
        const float* r2 = inputs + (size_t)(blk * SPB + tid) * NCLS;
        const float* ch = r2 + (t & ~3);          // 16B-aligned 4-chunk
        const float e0 = ch[0], e1 = ch[1], e2 = ch[2], e3 = ch[3];
        const int i = t & 3;
        const float pA0 = (i & 2) ? e2 : e0;      // pair containing t
        const float pA1 = (i & 2) ? e3 : e1;
        const float pB0 = (i & 2) ? e0 : e2;      // the other pair
        const float pB1 = (i & 2) ? e1 : e3;
        const float s0  = (i & 1) ? pA1 : pA0;    // p[t]
        sib[0] = (i & 1) ? pA0 : pA1;             // p[t^1]
        sib[1] = pB0 + pB1;

        const float* wrow = weights + (size_t)t * NEDGE;
        float loss = 0.0f;
        float sj = s0;
#pragma unroll
        for (int j = 0; j < NEDGE; ++j) {
            const float sn   = sj + sib[j];       // s_{j+1} (den)
            const float term = (sj != 0.0f) ? -logf(sj / sn) : 0.0f;
            loss += wrow[j] * term;
            sj = sn;
        }
        red[tid] = loss;
    }
    __syncthreads();

    if (tid == 0) {
        float acc = 0.0f;
#pragma unroll
        for (int k = 0; k < SPB; ++k) acc += red[k];
        partial[blk] = acc;
    }
}

// Fixed-order final reduction -> deterministic output across graph replays.
__global__ __launch_bounds__(64) void hll_reduce(const float* __restrict__ partial,
                                                 float* __restrict__ out)
{
    __shared__ float s[64];
    const int tid = threadIdx.x;
    float acc = 0.0f;
    for (int k = tid; k < NBLK; k += 64) acc += partial[k];
    s[tid] = acc;
    __syncthreads();
    if (tid == 0) {
        float tot = 0.0f;
#pragma unroll
        for (int k = 0; k < 64; ++k) tot += s[k];
        out[0] = tot / (float)BATCH;
    }
}

extern "C" void kernel_launch(void* const* d_in, const int* in_sizes, int n_in,
                              void* d_out, int out_size, void* d_ws, size_t ws_size,
                              hipStream_t stream)
{
    const float* inputs  = (const float*)d_in[0];   // [B, C] f32
    const int*   target  = (const int*)  d_in[1];   // [B] i32
    // d_in[2] onehot_num, d_in[3] onehot_den: implicit in the tree structure.
    const float* weights = (const float*)d_in[4];   // [C, E] f32

    float* partial = (float*)d_ws;                  // NBLK floats of scratch

    hll_main  <<<NBLK, 256, 0, stream>>>(inputs, target, weights, partial);
    hll_reduce<<<1,    64,  0, stream>>>(partial, (float*)d_out);
}